// SE3PointTwoLayerKernel_73031623901621
// MI455X (gfx1250) — compile-verified
//
#include <hip/hip_runtime.h>
#include <math.h>

// ---------------------------------------------------------------------------
// SE(3) two-layer point kernel for gfx1250 (MI455X).
//  - fp32 WMMA (v_wmma_f32_16x16x4_f32) for both MLP layers
//  - path-templated inner loops: all shape params compile-time
//  - D->B reshape of the hidden layer via __shfl_xor(16) (no LDS round trip)
//  - LDS staging of Y (sph. harmonics), windows, Q.Y contractions
//  - NT stores for the 340MB write-once output stream
// ---------------------------------------------------------------------------

typedef float v2f __attribute__((ext_vector_type(2)));
typedef float v8f __attribute__((ext_vector_type(8)));

constexpr int ODIM = 36;          // 4*1 + 4*3 + 4*5

// ---- path tables (RS_OUT outer, RS_IN inner; l values 0,1,2) --------------
constexpr int PLO[9]   = {0,0,0,1,1,1,2,2,2};
constexpr int PLI[9]   = {0,1,2,0,1,2,0,1,2};
constexpr int LENJS[9] = {1,1,1,1,3,3,1,3,5};   // #J per path (= #K chunks)
constexpr int J0TAB[9] = {0,1,2,1,0,1,2,1,0};   // first J (Js contiguous)
constexpr int DOUTT[9] = {1,1,1,3,3,3,5,5,5};
constexpr int DINT[9]  = {1,3,5,1,3,5,1,3,5};
constexpr int BEL[9]   = {4,4,4,4,12,12,4,12,20};           // 4*lenJs
constexpr int W0OFF[9] = {0,40,80,120,160,280,400,440,560}; // cum 10*b_el
constexpr int A0OFF[9] = {0,64,128,192,256,448,640,704,896};// cum 16*b_el
constexpr int QOFF[9][5] = {
  {0,0,0,0,0},{1,0,0,0,0},{10,0,0,0,0},{35,0,0,0,0},
  {44,53,80,0,0},{125,170,245,0,0},{350,0,0,0,0},
  {375,420,495,0,0},{600,625,700,825,1000}};   // float offsets of Q tables
constexpr int OBASE[3] = {0,4,16};              // out row/col base per l block
constexpr int YOFF[5]  = {0,1,4,9,16};          // Y row offset per J

// workspace layout (floats)
constexpr int WS_Q  = 0;      // 1225 (pad 1232)
constexpr int WS_A0 = 1232;   // 1216
constexpr int WS_A1 = 2448;   // 9*192
constexpr int WS_B0 = 4176;   // 9*16 (padded biases_0)
constexpr int WS_B1 = 4320;   // 9*16

// segments: (path, dodi_begin, dodi_count) with count*lenJs <= 27 LDS rows
constexpr int NSEG = 15;
constexpr int SEGP [NSEG] = {0,1,2,3,4,5,5,6,7,7,8,8,8,8,8};
constexpr int SEGD0[NSEG] = {0,0,0,0,0,0,9,0,0,9,0,5,10,15,20};
constexpr int SEGND[NSEG] = {1,3,5,3,9,9,6,5,9,6,5,5,5,5,5};

// ===========================================================================
// Init kernel: CG / Q tables (double), pre-scaled padded weight operands.
// ===========================================================================
__device__ double dfact(int n) { double r = 1.0; for (int i = 2; i <= n; ++i) r *= (double)i; return r; }

__device__ double cgc(int l1,int m1,int l2,int m2,int l3,int m3) {
  if (m1 + m2 != m3) return 0.0;
  double pre = sqrt((2.0*l3+1.0) * dfact(l3+l1-l2) * dfact(l3-l1+l2) * dfact(l1+l2-l3) / dfact(l1+l2+l3+1));
  pre *= sqrt(dfact(l3+m3)*dfact(l3-m3)*dfact(l1-m1)*dfact(l1+m1)*dfact(l2-m2)*dfact(l2+m2));
  double s = 0.0;
  for (int k = 0; k <= l1+l2-l3; ++k) {
    int d0=k, d1=l1+l2-l3-k, d2=l1-m1-k, d3=l2+m2-k, d4=l3-l2+m1+k, d5=l3-l1-m2+k;
    if (d0<0||d1<0||d2<0||d3<0||d4<0||d5<0) continue;
    double t = 1.0/(dfact(d0)*dfact(d1)*dfact(d2)*dfact(d3)*dfact(d4)*dfact(d5));
    s += (k & 1) ? -t : t;
  }
  return pre * s;
}

__device__ void buildU(int l, double (*Ure)[9], double (*Uim)[9]) {
  int d = 2*l + 1;
  for (int i = 0; i < d; ++i) for (int j = 0; j < d; ++j) { Ure[i][j]=0.0; Uim[i][j]=0.0; }
  Ure[l][l] = 1.0;
  const double is2 = 0.7071067811865476;
  for (int m = 1; m <= l; ++m) {
    double sg = (m & 1) ? -1.0 : 1.0;
    Ure[l+m][l+m] =  sg*is2;
    Ure[l+m][l-m] =  is2;
    Uim[l-m][l+m] = -sg*is2;
    Uim[l-m][l-m] =  is2;
  }
}

__device__ void computeQ(int p, int ji, float* dst) {
  int J = J0TAB[p] + ji, li = PLI[p], lo = PLO[p];
  int dO = 2*lo+1, dI = 2*li+1, dJ = 2*J+1;
  double CG[5][5][9];
  for (int a = 0; a < dO; ++a)
    for (int b = 0; b < dI; ++b)
      for (int c = 0; c < dJ; ++c)
        CG[a][b][c] = cgc(li, b-li, J, c-J, lo, a-lo);
  double Uo_re[9][9], Uo_im[9][9], Ui_re[9][9], Ui_im[9][9], UJ_re[9][9], UJ_im[9][9];
  buildU(lo, Uo_re, Uo_im); buildU(li, Ui_re, Ui_im); buildU(J, UJ_re, UJ_im);
  double Tre[5][5][9], Tim[5][5][9];
  double sr = 0.0, si = 0.0;
  for (int a = 0; a < dO; ++a)
    for (int b = 0; b < dI; ++b)
      for (int c = 0; c < dJ; ++c) {
        double re = 0.0, im = 0.0;
        for (int pp = 0; pp < dO; ++pp) {
          double ore = Uo_re[a][pp], oim = Uo_im[a][pp];
          if (ore == 0.0 && oim == 0.0) continue;
          for (int q = 0; q < dI; ++q) {
            double ire = Ui_re[b][q], iim = -Ui_im[b][q];     // conj
            if (ire == 0.0 && iim == 0.0) continue;
            double m1re = ore*ire - oim*iim;
            double m1im = ore*iim + oim*ire;
            for (int r = 0; r < dJ; ++r) {
              double g = CG[pp][q][r];
              if (g == 0.0) continue;
              double jre = UJ_re[c][r], jim = -UJ_im[c][r];   // conj
              re += (m1re*jre - m1im*jim) * g;
              im += (m1re*jim + m1im*jre) * g;
            }
          }
        }
        Tre[a][b][c] = re; Tim[a][b][c] = im;
        sr += fabs(re); si += fabs(im);
      }
  bool useRe = (sr >= si);
  for (int a = 0; a < dO; ++a)
    for (int b = 0; b < dI; ++b)
      for (int c = 0; c < dJ; ++c)
        dst[(a*dI + b)*dJ + c] = (float)(useRe ? Tre[a][b][c] : Tim[a][b][c]);
}

__global__ void __launch_bounds__(256)
se3_init(const float* __restrict__ w0g, const float* __restrict__ b0g,
         const float* __restrict__ w1g, const float* __restrict__ b1g,
         float* __restrict__ ws) {
  int t = threadIdx.x;
  // ---- Q tables: one thread per (path, ji) table ----
  if (t < 19) {
    int idx = t, p = -1, ji = 0;
    for (int pp = 0; pp < 9; ++pp) {
      if (p < 0) { if (idx < LENJS[pp]) { p = pp; ji = idx; } else idx -= LENJS[pp]; }
    }
    computeQ(p, ji, ws + WS_Q + QOFF[p][ji]);
  }
  // ---- A0 operand: layout ((chunk*2 + half)*16 + row)*2 + j,  k = 4c+2*half+j
  for (int i = t; i < 1216; i += 256) {
    int p = 0, off = i;
    while (off >= 16*BEL[p]) { off -= 16*BEL[p]; ++p; }
    int j = off & 1, rh = off >> 1, row = rh & 15, ch = rh >> 4;
    int c = ch >> 1, hf = ch & 1;
    int k = 4*c + 2*hf + j;
    float fv0 = sqrtf((float)DOUTT[p] / (10.0f * (float)BEL[p] * 9.0f));
    float v = 0.0f;
    if (row < 10 && k < BEL[p]) v = w0g[W0OFF[p] + row*BEL[p] + k] * fv0;
    ws[WS_A0 + A0OFF[p] + off] = v;
  }
  // ---- A1 operand: 16 x 12 (HIDDEN=10 padded), 3 chunks, same layout
  for (int i = t; i < 9*192; i += 256) {
    int p = i / 192, off = i - p*192;
    int j = off & 1, rh = off >> 1, row = rh & 15, ch = rh >> 4;
    int c = ch >> 1, hf = ch & 1;
    int k = 4*c + 2*hf + j;
    float fv1 = sqrtf((float)DOUTT[p] / 360.0f);
    float v = 0.0f;
    if (k < 10) v = w1g[160*p + row*10 + k] * fv1;
    ws[WS_A1 + i] = v;
  }
  // ---- padded biases ----
  for (int i = t; i < 144; i += 256) {
    int p = i >> 4, row = i & 15;
    ws[WS_B0 + i] = (row < 10) ? b0g[10*p + row] : 0.0f;
    ws[WS_B1 + i] = b1g[i];
  }
}

// ===========================================================================
// Path-templated wave worker: everything shape-dependent is compile-time.
// ===========================================================================
template <int P>
__device__ __forceinline__ void seg_work(int d0, int nd, int n, int wave,
                                         int hf, int l15,
                                         const float* __restrict__ ws,
                                         float* __restrict__ out,
                                         const float* sW, const float* sK) {
  constexpr int L    = LENJS[P];          // #K chunks of layer 0 (b_el = 4L)
  constexpr int DIN  = DINT[P];
  constexpr int DOUT = DOUTT[P];
  constexpr int OB   = OBASE[PLO[P]];
  constexpr int IB   = OBASE[PLI[P]];

  // ---- hoisted per-path operands in wave32 WMMA A / C-bias layouts --------
  float a0v[2*L], a1v[6], b0v[8], b1v[8];
  {
    const float* a0p = ws + WS_A0 + A0OFF[P];
#pragma unroll
    for (int c = 0; c < L; ++c) {
      int base = ((c*2 + hf)*16 + l15)*2;
      a0v[2*c]   = a0p[base];
      a0v[2*c+1] = a0p[base+1];
    }
    const float* a1p = ws + WS_A1 + 192*P;
#pragma unroll
    for (int c = 0; c < 3; ++c) {
      int base = ((c*2 + hf)*16 + l15)*2;
      a1v[2*c]   = a1p[base];
      a1v[2*c+1] = a1p[base+1];
    }
#pragma unroll
    for (int v = 0; v < 8; ++v) {
      b0v[v] = ws[WS_B0 + 16*P + hf*8 + v];
      b1v[v] = ws[WS_B1 + 16*P + hf*8 + v];
    }
  }

  const bool H = (hf != 0);
  const int totalU = nd * 16;
  for (int u = wave; u < totalU; u += 8) {
    const int tile = u & 15;
    const int dd   = u >> 4;
    const int dodi = d0 + dd;
    const int colN = tile*16 + l15;        // this lane's point-pair column

    // window values for this lane's column: registers, statically indexed
    float wv[4];
#pragma unroll
    for (int i = 0; i < 4; ++i) wv[i] = sW[i*256 + colN];

    // ---------- layer 0: h = w0 @ basis  (L chunks of K=4) ----------
    v8f acc = {0.f,0.f,0.f,0.f,0.f,0.f,0.f,0.f};
#pragma unroll
    for (int c = 0; c < L; ++c) {
      float bj[2];
#pragma unroll
      for (int j = 0; j < 2; ++j) {
        constexpr int CC = 0;  (void)CC;
        const int kL = 4*c + j;            // lane-half 0 holds K rows 4c,4c+1
        const int kH = 4*c + 2 + j;        // lane-half 1 holds K rows 4c+2,4c+3
        const int wiL = kL / L, jiL = kL % L;   // basis idx = wi*L + ji
        const int wiH = kH / L, jiH = kH % L;
        float wsel = H ? wv[wiH] : wv[wiL];
        int   row  = H ? (dd*L + jiH) : (dd*L + jiL);
        bj[j] = wsel * sK[row*256 + colN];
      }
      v2f A = {a0v[2*c], a0v[2*c+1]};
      v2f B = {bj[0], bj[1]};
      acc = __builtin_amdgcn_wmma_f32_16x16x4_f32(false, A, false, B,
                                                  (short)0, acc, false, false);
    }

    // ---------- bias + relu; D->B reshape via half-wave swap ----------
    float hv[8];
#pragma unroll
    for (int v = 0; v < 8; ++v) hv[v] = fmaxf(acc[v] + b0v[v], 0.0f);
    // D layout: vgpr v = row v (lanes<16) / row v+8 (lanes>=16), col = lane&15.
    // Layer-1 B needs rows 4c+2hf, +1 at own col: self or lane^16.
    float s0 = __shfl_xor(hv[0], 16), s1 = __shfl_xor(hv[1], 16);
    float s2 = __shfl_xor(hv[2], 16), s3 = __shfl_xor(hv[3], 16);
    float s6 = __shfl_xor(hv[6], 16), s7 = __shfl_xor(hv[7], 16);
    v2f B0 = { H ? s2    : hv[0], H ? s3    : hv[1] };  // rows {0,1}/{2,3}
    v2f B1 = { H ? s6    : hv[4], H ? s7    : hv[5] };  // rows {4,5}/{6,7}
    v2f B2 = { H ? hv[2] : s0,    H ? hv[3] : s1    };  // rows {8,9}/{10,11}

    // ---------- layer 1: ker = w1 @ h  (K=12, rows 10,11 are zero) ----------
    v8f acc2 = {0.f,0.f,0.f,0.f,0.f,0.f,0.f,0.f};
    {
      v2f A0c = {a1v[0], a1v[1]};
      v2f A1c = {a1v[2], a1v[3]};
      v2f A2c = {a1v[4], a1v[5]};
      acc2 = __builtin_amdgcn_wmma_f32_16x16x4_f32(false, A0c, false, B0,
                                                   (short)0, acc2, false, false);
      acc2 = __builtin_amdgcn_wmma_f32_16x16x4_f32(false, A1c, false, B1,
                                                   (short)0, acc2, false, false);
      acc2 = __builtin_amdgcn_wmma_f32_16x16x4_f32(false, A2c, false, B2,
                                                   (short)0, acc2, false, false);
    }

    // ---------- bias + scatter (NT stores; output is write-once) ----------
    const int dO_ = dodi / DIN;            // constant divisor
    const int dI_ = dodi - dO_*DIN;
    const int colpart = n*256 + tile*16 + l15;
#pragma unroll
    for (int v = 0; v < 8; ++v) {
      int rr = v + hf*8;                   // D row = m_out*4 + m_in
      int mo = rr >> 2, mi = rr & 3;
      int orow = OB + mo*DOUT + dO_;
      int ocol = IB + mi*DIN  + dI_;
      __builtin_nontemporal_store(acc2[v] + b1v[v],
                                  out + ((orow*ODIM + ocol) << 16) + colpart);
    }
  }
}

// ===========================================================================
// Main kernel: 1 workgroup per row n; 256 threads; 8 waves do WMMA tiles.
// ===========================================================================
__global__ void __launch_bounds__(256)
se3_main(const float* __restrict__ diff, const float* __restrict__ ws,
         float* __restrict__ out) {
  __shared__ float sY[25*256];   // spherical harmonics rows
  __shared__ float sW[4*256];    // radial windows
  __shared__ float sK[27*256];   // staged Q.Y contractions (per segment)

  const int t = threadIdx.x;
  const int n = blockIdx.x;
  const int wave = t >> 5, lane = t & 31;
  const int hf = lane >> 4, l15 = lane & 15;

  __builtin_prefetch(ws + WS_A0, 0, 1);   // global_prefetch_b8 of weight operands

  // ---------------- stage 1: harmonics + windows (thread t owns col m=t) ---
  {
    float dx = diff[(n*256 + t)*3 + 0];
    float dy = diff[(n*256 + t)*3 + 1];
    float dz = diff[(n*256 + t)*3 + 2];
    float r  = sqrtf(dx*dx + dy*dy + dz*dz);
    float rr = fmaxf(r, 1e-8f);
    float x = dx/rr, y = dy/rr, z = dz/rr;

    float Aa[5], Bb[5];
    Aa[0] = 1.0f; Bb[0] = 0.0f;
#pragma unroll
    for (int m = 1; m <= 4; ++m) { Aa[m] = x*Aa[m-1] - y*Bb[m-1]; Bb[m] = x*Bb[m-1] + y*Aa[m-1]; }

    float P[5][5];
    P[0][0]=1.0f; P[1][1]=1.0f; P[2][2]=3.0f; P[3][3]=15.0f; P[4][4]=105.0f;
#pragma unroll
    for (int m = 0; m <= 4; ++m) {
      if (m + 1 <= 4) P[m+1][m] = (float)(2*m+1) * z * P[m][m];
#pragma unroll
      for (int l = m+2; l <= 4; ++l)
        P[l][m] = ((float)(2*l-1)*z*P[l-1][m] - (float)(l-1+m)*P[l-2][m]) * (1.0f/(float)(l-m));
    }

    constexpr float FACT[9] = {1,1,2,6,24,120,720,5040,40320};
    const float INV4PI = 0.07957747154594767f;
    const float SQ2    = 1.41421356237309505f;
    int idx = 0;
#pragma unroll
    for (int l = 0; l <= 4; ++l) {
#pragma unroll
      for (int mm = -l; mm <= l; ++mm) {
        int am = mm < 0 ? -mm : mm;
        float Nc = sqrtf((float)(2*l+1) * INV4PI * FACT[l-am] / FACT[l+am]);
        float v;
        if (mm == 0)      v = Nc * P[l][0];
        else if (mm > 0)  v = SQ2 * Nc * P[l][am] * Aa[am];
        else              v = SQ2 * Nc * P[l][am] * Bb[am];
        sY[idx*256 + t] = v; ++idx;
      }
    }
    const float RADII[4] = {0.5f, 1.0f, 1.5f, 2.0f};
    const float INVS = 1.0f/0.6f;
    const float NORM = 0.6649038006690546f; // 1/(0.6*sqrt(2*pi))
#pragma unroll
    for (int i = 0; i < 4; ++i) {
      float u = (r - RADII[i]) * INVS;
      sW[i*256 + t] = __expf(-0.5f*u*u) * NORM;
    }
  }
  __syncthreads();

  // ---------------- segment loop -------------------------------------------
  for (int s = 0; s < NSEG; ++s) {
    __syncthreads();                       // previous segment's sK reads done
    const int p  = SEGP[s];
    const int d0 = SEGD0[s];
    const int nd = SEGND[s];
    const int L  = LENJS[p];
    const int J0p = J0TAB[p];

    // ---- stage K = Q . Y  into LDS (thread t owns col t) ----
    {
      int rowi = 0;
      for (int dd = 0; dd < nd; ++dd) {
        int dodi = d0 + dd;
        for (int ji = 0; ji < L; ++ji, ++rowi) {
          int J = J0p + ji;
          const float* q = ws + WS_Q + QOFF[p][ji] + dodi*(2*J+1);
          int yb = YOFF[J];
          float acc = 0.0f;
          for (int c = 0; c <= 2*J; ++c) acc += q[c] * sY[(yb+c)*256 + t];
          sK[rowi*256 + t] = acc;
        }
      }
    }
    __syncthreads();

    switch (p) {
      case 0: seg_work<0>(d0, nd, n, wave, hf, l15, ws, out, sW, sK); break;
      case 1: seg_work<1>(d0, nd, n, wave, hf, l15, ws, out, sW, sK); break;
      case 2: seg_work<2>(d0, nd, n, wave, hf, l15, ws, out, sW, sK); break;
      case 3: seg_work<3>(d0, nd, n, wave, hf, l15, ws, out, sW, sK); break;
      case 4: seg_work<4>(d0, nd, n, wave, hf, l15, ws, out, sW, sK); break;
      case 5: seg_work<5>(d0, nd, n, wave, hf, l15, ws, out, sW, sK); break;
      case 6: seg_work<6>(d0, nd, n, wave, hf, l15, ws, out, sW, sK); break;
      case 7: seg_work<7>(d0, nd, n, wave, hf, l15, ws, out, sW, sK); break;
      default: seg_work<8>(d0, nd, n, wave, hf, l15, ws, out, sW, sK); break;
    }
  }
}

// ===========================================================================
extern "C" void kernel_launch(void* const* d_in, const int* in_sizes, int n_in,
                              void* d_out, int out_size, void* d_ws, size_t ws_size,
                              hipStream_t stream) {
  (void)in_sizes; (void)n_in; (void)out_size; (void)ws_size;
  const float* diff = (const float*)d_in[0];
  const float* w0   = (const float*)d_in[1];
  const float* b0   = (const float*)d_in[2];
  const float* w1   = (const float*)d_in[3];
  const float* b1   = (const float*)d_in[4];
  float* ws  = (float*)d_ws;
  float* out = (float*)d_out;

  se3_init<<<dim3(1), dim3(256), 0, stream>>>(w0, b0, w1, b1, ws);
  se3_main<<<dim3(256), dim3(256), 0, stream>>>(diff, ws, out);
}